// SuperWarpF_35407710388377
// MI455X (gfx1250) — compile-verified
//
#include <hip/hip_runtime.h>
#include <hip/hip_bf16.h>

typedef __attribute__((ext_vector_type(2))) float v2f;
typedef __attribute__((ext_vector_type(8))) float v8f;

constexpr int kH = 512;
constexpr int kW = 512;
constexpr int kCIN = 64;
constexpr int kCOUT = 3;
constexpr int kXTILE = 128;   // output pixels per workgroup strip
constexpr int kXLDS = 130;    // staged cols (halo of 1 each side)
constexpr int kCCH = 32;      // channels staged per chunk
constexpr int kCPAD = 34;     // channel-dim pad (even -> 8B-aligned v2f, bank friendly)

// ---------------- Phase 1: 3x3 conv (64 -> 3) via V_WMMA_F32_16X16X4_F32 ----
__global__ __launch_bounds__(256) void conv3x3_wmma_kernel(
    const float* __restrict__ feat, const float* __restrict__ wgt,
    const float* __restrict__ bias, float* __restrict__ p)
{
  __shared__ __align__(16) float ldsF[3 * kXLDS * kCPAD];   // [row][xi][c]
  __shared__ __align__(16) float ldsW[kCOUT * kCIN * 9];    // same layout as global

  const int tid = threadIdx.x;
  const int x0  = blockIdx.x * kXTILE;
  const int y   = blockIdx.y;
  const int b   = blockIdx.z;

  for (int i = tid; i < kCOUT * kCIN * 9; i += 256) ldsW[i] = wgt[i];

  const int wv   = tid >> 5;       // wave 0..7 -> 16-px fragment
  const int lane = tid & 31;
  const int m    = lane & 15;      // A-matrix row (pixel)
  const int half = lane >> 4;      // 0: K=0,1   1: K=2,3
  const int n    = m;              // B/D column (output channel)
  const float bmask = (n < kCOUT) ? 1.0f : 0.0f;
  const int   nc    = (n < kCOUT) ? n : 0;

  v8f acc = {};

  for (int cc = 0; cc < kCIN; cc += kCCH) {
    // ---- stage feat chunk: coalesced global reads, channel-fastest LDS ----
    for (int i = tid; i < kCCH * 3 * kXLDS; i += 256) {
      int cl  = i / (3 * kXLDS);
      int rem = i - cl * (3 * kXLDS);
      int r   = rem / kXLDS;
      int xi  = rem - r * kXLDS;
      int yg  = y + r - 1;
      int xg  = x0 - 1 + xi;
      float v = 0.0f;
      if (yg >= 0 && yg < kH && xg >= 0 && xg < kW)
        v = feat[(((size_t)b * kCIN + (cc + cl)) * kH + yg) * kW + xg];
      ldsF[(r * kXLDS + xi) * kCPAD + cl] = v;
    }
    __syncthreads();

    // ---- accumulate: 9 taps x (32 channels / 4) WMMAs ----
    for (int ky = 0; ky < 3; ++ky) {
      for (int kx = 0; kx < 3; ++kx) {
        const int xi = wv * 16 + m + kx;                 // 0..129
        const float* arow = &ldsF[(ky * kXLDS + xi) * kCPAD];
        const int widx0 = (nc * kCIN + cc) * 9 + ky * 3 + kx;
        #pragma unroll
        for (int c0 = 0; c0 < kCCH; c0 += 4) {
          // A (16x4 f32): lanes 0-15 -> K=c0,c0+1 ; lanes 16-31 -> K=c0+2,c0+3
          v2f a = *(const v2f*)&arow[c0 + 2 * half];
          // B (4x16 f32): same K split, column = n; zero-mask n >= 3
          v2f bb;
          bb.x = ldsW[widx0 + (c0 + 2 * half) * 9] * bmask;
          bb.y = ldsW[widx0 + (c0 + 2 * half + 1) * 9] * bmask;
          acc = __builtin_amdgcn_wmma_f32_16x16x4_f32(
              false, a, false, bb, (short)0, acc, false, false);
        }
      }
    }
    __syncthreads();
  }

  // ---- D layout: lane L -> N=L&15, rows M = 8*(L>>4)+r ----
  if (n < kCOUT) {
    const float bs = bias[n];
    const int xb = x0 + wv * 16 + half * 8;
    const size_t base = (((size_t)b * kCOUT + n) * kH + y) * kW;
    #pragma unroll
    for (int r = 0; r < 8; ++r)
      p[base + xb + r] = acc[r] + bs;
  }
}

// ---------------- Phase 2: bicubic homography warp ---------------------------
__device__ __forceinline__ float cubicw(float t) {
  t = fabsf(t);
  float t2 = t * t, t3 = t2 * t;
  const float a = -0.5f;
  float w_in  = (a + 2.0f) * t3 - (a + 3.0f) * t2 + 1.0f;
  float w_out = a * t3 - 5.0f * a * t2 + 8.0f * a * t - 4.0f * a;
  return t <= 1.0f ? w_in : (t < 2.0f ? w_out : 0.0f);
}

__global__ __launch_bounds__(256) void warp_bicubic_kernel(
    const float* __restrict__ p, const float* __restrict__ mM,
    float* __restrict__ out)
{
  int idx = blockIdx.x * 256 + threadIdx.x;
  if (idx >= kH * kW) return;
  int x = idx & (kW - 1);
  int y = idx >> 9;

  // 3x3 inverse (adjugate / det)
  float m00=mM[0], m01=mM[1], m02=mM[2];
  float m10=mM[3], m11=mM[4], m12=mM[5];
  float m20=mM[6], m21=mM[7], m22=mM[8];
  float c00 =  (m11*m22 - m12*m21);
  float c10 = -(m10*m22 - m12*m20);
  float c20 =  (m10*m21 - m11*m20);
  float det = m00*c00 + m01*c10 + m02*c20;
  float id = 1.0f / det;
  float i00 = c00*id, i01 = -(m01*m22 - m02*m21)*id, i02 =  (m01*m12 - m02*m11)*id;
  float i10 = c10*id, i11 =  (m00*m22 - m02*m20)*id, i12 = -(m00*m12 - m02*m10)*id;
  float i20 = c20*id, i21 = -(m00*m21 - m01*m20)*id, i22 =  (m00*m11 - m01*m10)*id;

  float X = (float)x, Y = (float)y;
  float s0 = i00*X + i01*Y + i02;
  float s1 = i10*X + i11*Y + i12;
  float s2 = i20*X + i21*Y + i22;
  float sx = s0 / s2, sy = s1 / s2;
  bool valid = (sx >= 0.f) && (sx <= (float)(kW - 1)) &&
               (sy >= 0.f) && (sy <= (float)(kH - 1));

  float x0f = floorf(sx), y0f = floorf(sy);
  float fx = sx - x0f, fy = sy - y0f;

  float wx[4], wy[4];
  int ix[4], iy[4];
  #pragma unroll
  for (int a2 = 0; a2 < 4; ++a2) {
    float off = (float)(a2 - 1);
    wx[a2] = cubicw(fx - off);
    wy[a2] = cubicw(fy - off);
    int xx2 = (int)x0f + (a2 - 1);
    int yy2 = (int)y0f + (a2 - 1);
    ix[a2] = min(max(xx2, 0), kW - 1);
    iy[a2] = min(max(yy2, 0), kH - 1);
  }

  #pragma unroll
  for (int b = 0; b < 2; ++b) {
    #pragma unroll
    for (int c = 0; c < kCOUT; ++c) {
      const float* pc = p + (((size_t)b * kCOUT + c) * kH) * kW;
      float accv = 0.0f;
      #pragma unroll
      for (int a2 = 0; a2 < 4; ++a2) {
        const float* row = pc + (size_t)iy[a2] * kW;
        float rs = wx[0] * row[ix[0]] + wx[1] * row[ix[1]] +
                   wx[2] * row[ix[2]] + wx[3] * row[ix[3]];
        accv += wy[a2] * rs;
      }
      out[(((size_t)b * kCOUT + c) * kH + y) * kW + x] = valid ? accv : -255.0f;
    }
  }
}

// ---------------- Launch -----------------------------------------------------
extern "C" void kernel_launch(void* const* d_in, const int* in_sizes, int n_in,
                              void* d_out, int out_size, void* d_ws, size_t ws_size,
                              hipStream_t stream) {
  (void)in_sizes; (void)n_in; (void)out_size; (void)ws_size;
  const float* feat = (const float*)d_in[0];   // (2,64,512,512) f32
  const float* wgt  = (const float*)d_in[1];   // (3,64,3,3) f32
  const float* bias = (const float*)d_in[2];   // (3,) f32
  const float* mM   = (const float*)d_in[3];   // (3,3) f32
  float* out = (float*)d_out;                  // (2,3,512,512) f32
  float* p   = (float*)d_ws;                   // conv result: 2*3*512*512 f32

  dim3 g1(kW / kXTILE, kH, 2);                 // (4, 512, 2)
  conv3x3_wmma_kernel<<<g1, 256, 0, stream>>>(feat, wgt, bias, p);

  int npx = kH * kW;
  warp_bicubic_kernel<<<(npx + 255) / 256, 256, 0, stream>>>(p, mM, out);
}